// kernel_generated_3_21887153341409
// MI455X (gfx1250) — compile-verified
//
#include <hip/hip_runtime.h>

// y[i,o,n] = sum_{k,j} x[j, n', ((o-1)%14)+k-1] * W[i,k,1,j]
//          + sum_{k,j} x[j, n', o+k-1]          * W[i,k,0,j]
//   n' = (n-1)%14, w-index zero-padded outside [0,14).
// Single fp32 GEMM: M=196 rows (o,n), N=256 cols (i), K=768 (l,k,j),
// V_WMMA_F32_16X16X4_F32 (exact fp32), B staged in LDS, split-K x4 per block
// with a deterministic in-LDS reduction.

typedef __attribute__((ext_vector_type(2))) float v2f;
typedef __attribute__((ext_vector_type(4))) float v4f;
typedef __attribute__((ext_vector_type(8))) float v8f;

#define BCOL_STRIDE 772   // 768 + 4 pad floats -> conflict-free ds_load_b64

__global__ __launch_bounds__(128) void fused_conv_wmma_f32(
    const float* __restrict__ x,   // (128,14,14): x[j*196 + h*14 + w]
    const float* __restrict__ W,   // (256,3,2,128): W[i*768 + k*256 + l*128 + j]
    float* __restrict__ y)         // (256,14,14): y[i*196 + o*14 + n]
{
    __shared__ float smem[16 * BCOL_STRIDE + 3 * 8 * 32];
    float* Blds = smem;
    float* Red  = smem + 16 * BCOL_STRIDE;

    const int tid     = threadIdx.x;
    const int lane    = tid & 31;
    const int wave    = tid >> 5;        // 4 waves: split-K over K/4 each
    const int laneLow = lane & 15;
    const int kHalf   = lane >> 4;       // 0: K taps {0,1}; 1: K taps {2,3}

    const int nt = blockIdx.x;           // 0..15  N tile (i)
    const int mt = blockIdx.y;           // 0..12  M tile (o,n rows)

    // ---- Stage this block's 16 W columns (contiguous 48KB) into padded LDS ----
    {
        const float* src = W + nt * (16 * 768);
        #pragma unroll
        for (int it = 0; it < 24; ++it) {
            const int chunk  = tid + it * 128;        // 3072 float4 chunks total
            const int col    = chunk / 192;           // 192 chunks per column
            const int within = (chunk % 192) * 4;
            v4f v = *(const v4f*)(src + chunk * 4);
            *(v4f*)(Blds + col * BCOL_STRIDE + within) = v;
        }
    }
    __syncthreads();

    // ---- A-row mapping for this lane ----
    const int  mRow     = mt * 16 + laneLow;
    const bool rowValid = mRow < 196;
    const int  o        = rowValid ? (mRow / 14) : 0;
    const int  n        = rowValid ? (mRow % 14) : 0;
    const int  oEff1    = (o + 13) % 14;   // (o-1) mod 14  (l=1 term)
    const int  nEff     = (n + 13) % 14;   // (n-1) mod 14

    const float* bcol = Blds + laneLow * BCOL_STRIDE;
    const int    jt0  = wave * 8;          // this wave's j-step window per phase

    v8f acc0 = {};
    v8f acc1 = {};

    #pragma unroll
    for (int l2 = 1; l2 >= 0; --l2) {
        #pragma unroll
        for (int k = 0; k < 3; ++k) {
            const int  w      = (l2 ? oEff1 : o) + k - 1;      // zero-padded tap
            const bool aValid = rowValid && (w >= 0) && (w < 14);
            const float* xb = x + nEff * 14 + w;               // + j*196
            const float* wb = bcol + k * 256 + l2 * 128;       // + j  (LDS)

            v2f a[8], b[8];
            // batch global A loads (one clause, one wait) ...
            #pragma unroll
            for (int t = 0; t < 8; ++t) {
                const int j = 4 * (jt0 + t) + 2 * kHalf;
                a[t].x = aValid ? xb[j * 196]       : 0.0f;
                a[t].y = aValid ? xb[(j + 1) * 196] : 0.0f;
            }
            // ... batch LDS B loads ...
            #pragma unroll
            for (int t = 0; t < 8; ++t) {
                const int j = 4 * (jt0 + t) + 2 * kHalf;
                b[t] = *(const v2f*)(wb + j);
            }
            // ... then 8 WMMAs on two accumulator chains
            #pragma unroll
            for (int t = 0; t < 8; t += 2) {
                acc0 = __builtin_amdgcn_wmma_f32_16x16x4_f32(
                    false, a[t],     false, b[t],     (short)0, acc0, false, false);
                acc1 = __builtin_amdgcn_wmma_f32_16x16x4_f32(
                    false, a[t + 1], false, b[t + 1], (short)0, acc1, false, false);
            }
        }
    }

    // ---- Deterministic split-K reduction across the 4 waves ----
    v8f acc;
    #pragma unroll
    for (int vr = 0; vr < 8; ++vr) acc[vr] = acc0[vr] + acc1[vr];

    if (wave != 0) {
        #pragma unroll
        for (int vr = 0; vr < 8; ++vr)
            Red[((wave - 1) * 8 + vr) * 32 + lane] = acc[vr];
    }
    __syncthreads();

    if (wave == 0) {
        const int iOut = nt * 16 + laneLow;
        #pragma unroll
        for (int vr = 0; vr < 8; ++vr) {
            float s = acc[vr];
            s += Red[(0 * 8 + vr) * 32 + lane];
            s += Red[(1 * 8 + vr) * 32 + lane];
            s += Red[(2 * 8 + vr) * 32 + lane];
            // C/D layout: VGPR vr -> M = vr + 8*kHalf, N = laneLow
            const int R = mt * 16 + vr + 8 * kHalf;
            if (R < 196) {
                const int oo = R / 14;
                const int nn = R % 14;
                y[iOut * 196 + oo * 14 + nn] = s;
            }
        }
    }
}

extern "C" void kernel_launch(void* const* d_in, const int* in_sizes, int n_in,
                              void* d_out, int out_size, void* d_ws, size_t ws_size,
                              hipStream_t stream) {
    (void)in_sizes; (void)n_in; (void)out_size; (void)d_ws; (void)ws_size;
    const float* x = (const float*)d_in[0];   // (1,128,14,14) fp32
    const float* W = (const float*)d_in[1];   // (256,3,2,128) fp32
    float* y = (float*)d_out;                 // (1,256,14,14) fp32

    dim3 grid(16, 13);  // N tiles x M tiles; 4 split-K waves per block
    dim3 block(128);
    hipLaunchKernelGGL(fused_conv_wmma_f32, grid, block, 0, stream, x, W, y);
}